// InvertibleConv1x1SubblocksShuf_17961553232469
// MI455X (gfx1250) — compile-verified
//
#include <hip/hip_runtime.h>
#include <math.h>

typedef __attribute__((ext_vector_type(2))) float v2f;
typedef __attribute__((ext_vector_type(8))) float v8f;

#define HW        16384      // 128*128
#define C_BLK     32
#define NB        8
#define BATCH     16
#define PX_TILE   16
#define TILES_PER_WAVE 8                         // 128 pixels / wave strip
#define STRIPS_PER_BK  (HW / (PX_TILE * TILES_PER_WAVE))   // 128

// z[b,k,d,p] = sum_c w[k,d,c] * x[b,k,c,p]  via V_WMMA_F32_16X16X4_F32
__global__ __launch_bounds__(256)
void bd1x1_wmma_kernel(const float* __restrict__ x,
                       const float* __restrict__ w,
                       float* __restrict__ z) {
    const int lane      = threadIdx.x & 31;
    const int waveInBlk = threadIdx.x >> 5;
    const int gwave     = blockIdx.x * 8 + waveInBlk;

    const int strip = gwave & (STRIPS_PER_BK - 1);   // 0..127
    const int k     = (gwave >> 7) & (NB - 1);       // 0..7
    const int b     = gwave >> 10;                   // 0..15

    const int hi     = lane >> 4;     // 0: lanes 0-15, 1: lanes 16-31
    const int lolane = lane & 15;

    // ---- A fragments: W_k resident in registers for the whole strip ----
    // A 16x4 f32 layout: lane (lo) holds row M=lolane; v2f element j is
    // K = 4*kc + 2*hi + j.
    const float* wk = w + (size_t)k * (C_BLK * C_BLK);
    v2f a[2][8];
#pragma unroll
    for (int r = 0; r < 2; ++r) {
        const float* wrow = wk + (size_t)(r * 16 + lolane) * C_BLK + 2 * hi;
#pragma unroll
        for (int kc = 0; kc < 8; ++kc) {
            a[r][kc].x = wrow[kc * 4 + 0];   // contiguous pair -> b64 load
            a[r][kc].y = wrow[kc * 4 + 1];
        }
    }

    const size_t bkbase = (size_t)(b * NB + k) * C_BLK * HW;
    const float* xb = x + bkbase;
    float*       zb = z + bkbase;
    const int px0 = strip * (PX_TILE * TILES_PER_WAVE);

    for (int t = 0; t < TILES_PER_WAVE; ++t) {
        const int px = px0 + t * PX_TILE + lolane;

        // ---- B fragments: x[c, px] with c = 4*kc + 2*hi + j ----
        // each .x/.y load is 16 consecutive floats per half-wave (64B)
        v2f bf[8];
#pragma unroll
        for (int kc = 0; kc < 8; ++kc) {
            const int c0 = kc * 4 + 2 * hi;
            bf[kc].x = xb[(size_t)(c0 + 0) * HW + px];
            bf[kc].y = xb[(size_t)(c0 + 1) * HW + px];
        }

        v8f acc0 = {0.f, 0.f, 0.f, 0.f, 0.f, 0.f, 0.f, 0.f};
        v8f acc1 = {0.f, 0.f, 0.f, 0.f, 0.f, 0.f, 0.f, 0.f};
#pragma unroll
        for (int kc = 0; kc < 8; ++kc) {
            acc0 = __builtin_amdgcn_wmma_f32_16x16x4_f32(
                false, a[0][kc], false, bf[kc], (short)0, acc0, false, false);
            acc1 = __builtin_amdgcn_wmma_f32_16x16x4_f32(
                false, a[1][kc], false, bf[kc], (short)0, acc1, false, false);
        }

        // ---- store D: VGPR i holds row (i + 8*hi); rows 16..31 from acc1 ----
#pragma unroll
        for (int i = 0; i < 8; ++i) {
            zb[(size_t)(i + 8 * hi) * HW + px]      = acc0[i];
            zb[(size_t)(16 + i + 8 * hi) * HW + px] = acc1[i];
        }
    }
}

// slogdet of the eight 32x32 blocks + logdet update (tiny side kernel).
__global__ __launch_bounds__(32)
void logdet_kernel(const float* __restrict__ w,
                   const float* __restrict__ logdet_in,
                   float* __restrict__ out_logdet) {
    __shared__ float m[NB][C_BLK][C_BLK];
    __shared__ float partial[NB];

    for (int i = threadIdx.x; i < NB * C_BLK * C_BLK; i += 32)
        ((float*)m)[i] = w[i];
    __syncthreads();

    const int t = threadIdx.x;
    if (t < NB) {
        float ld = 0.f;
        for (int j = 0; j < C_BLK; ++j) {
            // partial pivot
            int piv = j;
            float pv = fabsf(m[t][j][j]);
            for (int i = j + 1; i < C_BLK; ++i) {
                float v = fabsf(m[t][i][j]);
                if (v > pv) { pv = v; piv = i; }
            }
            if (piv != j) {
                for (int c = j; c < C_BLK; ++c) {
                    float tmp = m[t][j][c];
                    m[t][j][c] = m[t][piv][c];
                    m[t][piv][c] = tmp;
                }
            }
            const float d = m[t][j][j];
            ld += logf(fabsf(d));
            const float inv = 1.0f / d;
            for (int i = j + 1; i < C_BLK; ++i) {
                const float f = m[t][i][j] * inv;
                for (int c = j + 1; c < C_BLK; ++c)
                    m[t][i][c] -= f * m[t][j][c];
            }
        }
        partial[t] = ld;
    }
    __syncthreads();

    if (t < BATCH) {
        float s = 0.f;
        for (int i = 0; i < NB; ++i) s += partial[i];
        out_logdet[t] = logdet_in[t] + s * (float)HW;
    }
}

extern "C" void kernel_launch(void* const* d_in, const int* in_sizes, int n_in,
                              void* d_out, int out_size, void* d_ws, size_t ws_size,
                              hipStream_t stream) {
    (void)in_sizes; (void)n_in; (void)out_size; (void)d_ws; (void)ws_size;
    const float* x      = (const float*)d_in[0];
    const float* w      = (const float*)d_in[1];
    const float* logdet = (const float*)d_in[2];

    float* z      = (float*)d_out;
    float* out_ld = z + (size_t)BATCH * (NB * C_BLK) * HW;  // after 16*256*128*128 floats

    const int totalWaves = BATCH * NB * STRIPS_PER_BK;      // 16384
    const int blocks     = totalWaves / 8;                  // 2048 blocks x 256 threads
    bd1x1_wmma_kernel<<<blocks, 256, 0, stream>>>(x, w, z);
    logdet_kernel<<<1, 32, 0, stream>>>(w, logdet, out_ld);
}